// GraphSAGELayer_3384434229646
// MI455X (gfx1250) — compile-verified
//
#include <hip/hip_runtime.h>

typedef __attribute__((ext_vector_type(2))) float v2f;
typedef __attribute__((ext_vector_type(8))) float v8f;

#define N_NODES 100000
#define N_EDGES 1600000
#define D 128

// ---------------- zero workspace (agg + deg) ----------------
__global__ void sage_zero_kernel(float* __restrict__ ws, long long n) {
  long long i = (long long)blockIdx.x * blockDim.x + threadIdx.x;
  long long stride = (long long)gridDim.x * blockDim.x;
  for (; i < n; i += stride) ws[i] = 0.0f;
}

// ---------------- edge scatter: agg[dst] += x[src], deg[dst] += 1 ----------------
// 32 threads per edge; each thread handles 4 consecutive floats (float4 load,
// 4 global f32 atomics). x and agg are both L2-resident (51.2 MB each).
__global__ void sage_edge_scatter_kernel(const float* __restrict__ x,
                                         const long long* __restrict__ src,
                                         const long long* __restrict__ dst,
                                         float* __restrict__ agg,
                                         float* __restrict__ deg) {
  long long t = (long long)blockIdx.x * blockDim.x + threadIdx.x;
  long long e = t >> 5;
  int c = (int)(t & 31);
  if (e >= N_EDGES) return;
  long long s = src[e];
  long long d = dst[e];
  const float4* xs = (const float4*)(x + s * D);
  float4 v = xs[c];
  float* o = agg + d * D + (long long)c * 4;
  atomicAdd(o + 0, v.x);
  atomicAdd(o + 1, v.y);
  atomicAdd(o + 2, v.z);
  atomicAdd(o + 3, v.w);
  if (c == 0) atomicAdd(deg + d, 1.0f);
}

// ---------------- agg /= max(deg, 1) ----------------
__global__ void sage_normalize_kernel(float* __restrict__ agg,
                                      const float* __restrict__ deg) {
  long long i = (long long)blockIdx.x * blockDim.x + threadIdx.x;
  if (i >= (long long)N_NODES * D) return;
  float dg = deg[i >> 7];
  agg[i] *= 1.0f / fmaxf(dg, 1.0f);
}

// ---------------- fused GEMM: relu(x@Ws + agg@Wn + bs + bn) ----------------
// One wave per 16x16 output tile; block = 16 rows x 128 cols (8 waves).
// K = 256 (x columns 0..127, agg columns 128..255), 4 per WMMA -> 64 WMMA/wave.
__global__ __launch_bounds__(256)
void sage_gemm_wmma_kernel(const float* __restrict__ x,
                           const float* __restrict__ agg,
                           const float* __restrict__ Wself,
                           const float* __restrict__ bself,
                           const float* __restrict__ Wneigh,
                           const float* __restrict__ bneigh,
                           float* __restrict__ out) {
  const int lane = threadIdx.x & 31;
  const int wave = threadIdx.x >> 5;     // 0..7 -> output column tile
  const int m0 = blockIdx.x * 16;        // output row tile
  const int n0 = wave * 16;
  const int half = lane >> 4;            // which half of the wave
  const int l = lane & 15;

  v8f acc = {};

  // A 16x4 f32 fragment: lane l (half 0) holds row m0+l, K = k0, k0+1;
  // lane 16+l (half 1) holds row m0+l, K = k0+2, k0+3.
  // B 4x16 f32 fragment mirrors: lane l holds col n0+l, same K split.
  const float* Ax = x + (long long)(m0 + l) * D;
  const float* Aa = agg + (long long)(m0 + l) * D;

#pragma unroll 8
  for (int k0 = 0; k0 < D; k0 += 4) {
    const int kk = k0 + 2 * half;
    v2f a = *(const v2f*)(Ax + kk);
    v2f b;
    b.x = Wself[(long long)kk * D + n0 + l];
    b.y = Wself[(long long)(kk + 1) * D + n0 + l];
    acc = __builtin_amdgcn_wmma_f32_16x16x4_f32(false, a, false, b,
                                                (short)0, acc, false, false);
  }
#pragma unroll 8
  for (int k0 = 0; k0 < D; k0 += 4) {
    const int kk = k0 + 2 * half;
    v2f a = *(const v2f*)(Aa + kk);
    v2f b;
    b.x = Wneigh[(long long)kk * D + n0 + l];
    b.y = Wneigh[(long long)(kk + 1) * D + n0 + l];
    acc = __builtin_amdgcn_wmma_f32_16x16x4_f32(false, a, false, b,
                                                (short)0, acc, false, false);
  }

  const float bias = bself[n0 + l] + bneigh[n0 + l];

  // C/D layout: VGPR i -> row m0 + i + 8*half, col n0 + l.
#pragma unroll
  for (int i = 0; i < 8; ++i) {
    const int row = m0 + i + 8 * half;
    float v = acc[i] + bias;
    out[(long long)row * D + (n0 + l)] = fmaxf(v, 0.0f);
  }
}

extern "C" void kernel_launch(void* const* d_in, const int* in_sizes, int n_in,
                              void* d_out, int out_size, void* d_ws, size_t ws_size,
                              hipStream_t stream) {
  (void)in_sizes; (void)n_in; (void)out_size; (void)ws_size;
  const float* x          = (const float*)d_in[0];
  const long long* ei     = (const long long*)d_in[1];  // int64 [2, N_EDGES]
  const float* Wself      = (const float*)d_in[2];
  const float* bself      = (const float*)d_in[3];
  const float* Wneigh     = (const float*)d_in[4];
  const float* bneigh     = (const float*)d_in[5];
  float* out = (float*)d_out;

  float* agg = (float*)d_ws;                        // N_NODES * D floats
  float* deg = agg + (size_t)N_NODES * D;           // N_NODES floats

  // 1) zero agg + deg
  const long long zn = (long long)N_NODES * D + N_NODES;
  sage_zero_kernel<<<2048, 256, 0, stream>>>(agg, zn);

  // 2) scatter-add over edges (32 threads/edge)
  const long long tot = (long long)N_EDGES * 32;
  const unsigned grid_e = (unsigned)((tot + 255) / 256);
  sage_edge_scatter_kernel<<<grid_e, 256, 0, stream>>>(x, ei, ei + N_EDGES, agg, deg);

  // 3) normalize by in-degree
  const long long ne = (long long)N_NODES * D;
  sage_normalize_kernel<<<(unsigned)((ne + 255) / 256), 256, 0, stream>>>(agg, deg);

  // 4) fused dual GEMM + bias + relu via f32 WMMA
  sage_gemm_wmma_kernel<<<N_NODES / 16, 256, 0, stream>>>(
      x, agg, Wself, bself, Wneigh, bneigh, out);
}